// CapsuleLinear_28552942584473
// MI455X (gfx1250) — compile-verified
//
#include <hip/hip_runtime.h>

typedef __attribute__((ext_vector_type(2))) float v2f;
typedef __attribute__((ext_vector_type(8))) float v8f;

#define B_SZ     64
#define IN_CAPS  1152
#define IN_LEN   8
#define OUT_CAPS 64
#define OUT_LEN  16
#define NTILES   (IN_CAPS / 16)        // 72 row-tiles of 16
#define TPB      256                   // 8 waves (wave32)
#define NWAVES   (TPB / 32)
#define TILES_PER_WAVE (NTILES / NWAVES) // 9
#define PSTR     18                    // LDS row stride (floats); 8*18 % 64 == 16 -> conflict-free C spill
#define NITER    3
#define EPS      1e-12f

// One workgroup handles one (b, o) pair: computes the 1152x16 fp32 priors tile
// with V_WMMA_F32_16X16X4_F32 (two K=4 steps for K=8), keeps it in LDS, then
// runs 3 k-means routing iterations entirely in LDS. Writes out[b,o,16] and
// probs[b,o,1152].
__global__ __launch_bounds__(TPB) void capsule_kmeans_kernel(
    const float* __restrict__ x,      // [B, IN_CAPS, IN_LEN]
    const float* __restrict__ w,      // [OUT_CAPS, OUT_LEN, IN_LEN]
    float* __restrict__ out,          // [B, OUT_CAPS, OUT_LEN]
    float* __restrict__ probs)        // [B, OUT_CAPS, IN_CAPS]
{
    extern __shared__ float smem[];
    float* sP    = smem;                      // priors, PSTR * IN_CAPS floats
    float* sE    = sP + PSTR * IN_CAPS;       // logits / exp scratch, IN_CAPS
    float* red   = sE + IN_CAPS;              // reduction scratch, TPB
    float* cent  = red + TPB;                 // centroid, 16
    float* centn = cent + 16;                 // normalized centroid, 16
    float* sc    = centn + 16;                // scalars, 4

    const int tid  = threadIdx.x;
    const int lane = tid & 31;
    const int wave = tid >> 5;
    const int hi   = lane >> 4;               // lane half (K-pair select)
    const int lm   = lane & 15;

    const int blk = blockIdx.x;               // 0 .. B*O-1
    const int b   = blk >> 6;
    const int o   = blk & 63;
    const float* __restrict__ xb = x + (size_t)b * IN_CAPS * IN_LEN;
    const float* __restrict__ wo = w + (size_t)o * OUT_LEN * IN_LEN;

    // ---- B operand (wT : 8 x 16), shared by all tiles ----
    // B 4x16 layout: VGPR0 = B[2*hi + 0, n=lm], VGPR1 = B[2*hi + 1, n=lm]
    // B[k, l] = weight[o, l, k] -> contiguous float2 at wo + l*8 + 2*hi
    v2f bm0, bm1;
    {
        const float* wp = wo + lm * IN_LEN + 2 * hi;
        bm0.x = wp[0]; bm0.y = wp[1];   // K = {0,1} or {2,3}
        bm1.x = wp[4]; bm1.y = wp[5];   // K = {4,5} or {6,7}
    }

    // ---- priors tiles via WMMA f32 16x16x4, spill C to LDS ----
    for (int t = 0; t < TILES_PER_WAVE; ++t) {
        const int tile  = wave * TILES_PER_WAVE + t;
        const int rbase = tile * 16;
        // A 16x4 layout: lanes 0-15 K={0,1}, lanes 16-31 K={2,3}; row = lm
        const float* xp = xb + (size_t)(rbase + lm) * IN_LEN + 2 * hi;
        v2f a0, a1;
        a0.x = xp[0]; a0.y = xp[1];
        a1.x = xp[4]; a1.y = xp[5];
        v8f c = {0.f, 0.f, 0.f, 0.f, 0.f, 0.f, 0.f, 0.f};
        c = __builtin_amdgcn_wmma_f32_16x16x4_f32(false, a0, false, bm0,
                                                  (short)0, c, false, false);
        c = __builtin_amdgcn_wmma_f32_16x16x4_f32(false, a1, false, bm1,
                                                  (short)0, c, false, false);
        // C layout: VGPR v -> row (v + 8*hi), col lm
        const int rw = rbase + 8 * hi;
#pragma unroll
        for (int v = 0; v < 8; ++v)
            sP[(rw + v) * PSTR + lm] = c[v];
    }
    __syncthreads();

    // ---- initial centroid = mean over n ----
    {
        const int l = tid & 15, g = tid >> 4;
        float acc = 0.f;
        for (int n = g; n < IN_CAPS; n += 16) acc += sP[n * PSTR + l];
        red[g * 16 + l] = acc;
    }
    __syncthreads();
    if (tid < 16) {
        float s = 0.f;
#pragma unroll
        for (int g = 0; g < 16; ++g) s += red[g * 16 + tid];
        cent[tid] = s * (1.0f / (float)IN_CAPS);
    }
    __syncthreads();

    // ---- routing iterations ----
    for (int it = 0; it < NITER; ++it) {
        // normalize centroid (F.normalize: v / max(||v||, eps))
        if (tid == 0) {
            float s = 0.f;
#pragma unroll
            for (int l = 0; l < 16; ++l) s += cent[l] * cent[l];
            sc[0] = 1.0f / fmaxf(sqrtf(s), EPS);
        }
        __syncthreads();
        if (tid < 16) centn[tid] = cent[tid] * sc[0];
        __syncthreads();

        // logits + block max
        float lmax = -3.4e38f;
        for (int n = tid; n < IN_CAPS; n += TPB) {
            float d = 0.f;
#pragma unroll
            for (int l = 0; l < 16; ++l) d += sP[n * PSTR + l] * centn[l];
            sE[n] = d;
            lmax = fmaxf(lmax, d);
        }
        red[tid] = lmax;
        __syncthreads();
        for (int s = TPB / 2; s > 0; s >>= 1) {
            if (tid < s) red[tid] = fmaxf(red[tid], red[tid + s]);
            __syncthreads();
        }
        const float gmax = red[0];
        __syncthreads();

        // exp + block sum
        float lsum = 0.f;
        for (int n = tid; n < IN_CAPS; n += TPB) {
            float e = __expf(sE[n] - gmax);
            sE[n] = e;
            lsum += e;
        }
        red[tid] = lsum;
        __syncthreads();
        for (int s = TPB / 2; s > 0; s >>= 1) {
            if (tid < s) red[tid] += red[tid + s];
            __syncthreads();
        }
        const float invsum = 1.0f / red[0];
        __syncthreads();

        // final-iteration probs output (softmax of last iteration)
        if (it == NITER - 1) {
            float* pb = probs + (size_t)blk * IN_CAPS;
            for (int n = tid; n < IN_CAPS; n += TPB)
                pb[n] = sE[n] * invsum;
        }

        // centroid update: cent[l] = (sum_n e[n] * priors[n,l]) * invsum
        {
            const int l = tid & 15, g = tid >> 4;
            float acc = 0.f;
            for (int n = g; n < IN_CAPS; n += 16) acc += sE[n] * sP[n * PSTR + l];
            red[g * 16 + l] = acc;
        }
        __syncthreads();
        if (tid < 16) {
            float s = 0.f;
#pragma unroll
            for (int g = 0; g < 16; ++g) s += red[g * 16 + tid];
            cent[tid] = s * invsum;
        }
        __syncthreads();
    }

    if (tid < 16)
        out[(size_t)blk * OUT_LEN + tid] = cent[tid];
}

extern "C" void kernel_launch(void* const* d_in, const int* in_sizes, int n_in,
                              void* d_out, int out_size, void* d_ws, size_t ws_size,
                              hipStream_t stream) {
    (void)in_sizes; (void)n_in; (void)d_ws; (void)ws_size; (void)out_size;
    const float* x = (const float*)d_in[0];   // [64, 1152, 8] fp32
    const float* w = (const float*)d_in[1];   // [64, 16, 8]   fp32
    float* out   = (float*)d_out;                                  // [64,64,16]
    float* probs = out + (size_t)B_SZ * OUT_CAPS * OUT_LEN;        // [64,64,1152]

    const size_t shmem =
        (size_t)(PSTR * IN_CAPS + IN_CAPS + TPB + 16 + 16 + 4) * sizeof(float);

    hipLaunchKernelGGL(capsule_kmeans_kernel,
                       dim3(B_SZ * OUT_CAPS), dim3(TPB), shmem, stream,
                       x, w, out, probs);
}